// RRN_40398462386652
// MI455X (gfx1250) — compile-verified
//
#include <hip/hip_runtime.h>
#include <hip/hip_bf16.h>

typedef __attribute__((ext_vector_type(2))) float v2f;
typedef __attribute__((ext_vector_type(8))) float v8f;

__device__ __forceinline__ float sigmoidf_(float x) { return 1.0f / (1.0f + __expf(-x)); }

__device__ __forceinline__ v8f wmma4(v2f a, v2f b, v8f c) {
  return __builtin_amdgcn_wmma_f32_16x16x4_f32(false, a, false, b, (short)0, c,
                                               false, false);
}

// ---------------------------------------------------------------------------
// Fused GEMM on f32 WMMA (V_WMMA_F32_16X16X4_F32):
//   C[M,N] = act( A1[M,K1] @ W1 + A2[M,K2] @ W2 + bias_scale*bias1 + bias2 )
// W layout: (K x N) row-major if !WTRANS, (N x K) row-major if WTRANS
// (WTRANS only applies to W1; the A2/W2 path is always non-transposed).
// Each wave owns a 16(M) x 32(N) C tile: one A fragment feeds two WMMAs.
// 8 waves / 256-thread block (wave32).
//
// Fragment layouts per CDNA5 ISA 7.12.2:
//   A 16x4 f32 (2 VGPRs): lane 0-15 = M, lane half selects K base (0 / 2).
//   B 4x16 mirrored (lane = N).  C/D: 8 VGPRs, VGPR r: lanes 0-15 ->
//   (M=r, N=lane), lanes 16-31 -> (M=8+r, N=lane-16).
// ---------------------------------------------------------------------------
template<bool WTRANS, bool HASA2, bool RELU>
__global__ void __launch_bounds__(256)
gemm_wmma_f32(const float* __restrict__ A1, int lda1,
              const float* __restrict__ W1, int ldw1, int K1,
              const float* __restrict__ A2, int lda2,
              const float* __restrict__ W2, int ldw2, int K2,
              const float* __restrict__ b1p, const float* __restrict__ b2p,
              float bias_scale,
              float* __restrict__ C, int ldc, int Mrows, int Ncols)
{
  const int wave = threadIdx.x >> 5;
  const int lane = threadIdx.x & 31;
  const int ntn  = Ncols >> 5;                 // 32-wide N tiles
  const int total = (Mrows >> 4) * ntn;
  const int tile = blockIdx.x * 8 + wave;
  if (tile >= total) return;                   // wave-uniform exit: EXEC all-ones in live waves
  const int mt = tile / ntn, nt = tile - mt * ntn;
  const int m0 = mt << 4, n0 = nt << 5;
  const int half = lane >> 4;                  // K-pair this lane holds (K+0/1 vs K+2/3)
  const int lp   = lane & 15;                  // M (A) / N (B, C)

  v8f acc0 = {}, acc1 = {};

  {
    const float* ap  = A1 + (size_t)(m0 + lp) * lda1 + half * 2;
    const float* wp0;
    const float* wp1;
    if (WTRANS) {
      wp0 = W1 + (size_t)(n0 + lp) * ldw1 + half * 2;
      wp1 = wp0 + (size_t)16 * ldw1;
    } else {
      wp0 = W1 + (size_t)(half * 2) * ldw1 + n0 + lp;
      wp1 = wp0 + 16;
    }
    #pragma unroll 4
    for (int k0 = 0; k0 < K1; k0 += 4) {
      v2f a, b0, b1;
      a.x = ap[0]; a.y = ap[1]; ap += 4;
      if (WTRANS) {
        b0.x = wp0[0]; b0.y = wp0[1]; wp0 += 4;
        b1.x = wp1[0]; b1.y = wp1[1]; wp1 += 4;
      } else {
        b0.x = wp0[0]; b0.y = wp0[ldw1];
        b1.x = wp1[0]; b1.y = wp1[ldw1];
        wp0 += (size_t)4 * ldw1; wp1 += (size_t)4 * ldw1;
      }
      acc0 = wmma4(a, b0, acc0);
      acc1 = wmma4(a, b1, acc1);
    }
  }

  if (HASA2) {
    const float* ap  = A2 + (size_t)(m0 + lp) * lda2 + half * 2;
    const float* wp0 = W2 + (size_t)(half * 2) * ldw2 + n0 + lp;
    const float* wp1 = wp0 + 16;
    #pragma unroll 4
    for (int k0 = 0; k0 < K2; k0 += 4) {
      v2f a, b0, b1;
      a.x = ap[0]; a.y = ap[1]; ap += 4;
      b0.x = wp0[0]; b0.y = wp0[ldw2];
      b1.x = wp1[0]; b1.y = wp1[ldw2];
      wp0 += (size_t)4 * ldw2; wp1 += (size_t)4 * ldw2;
      acc0 = wmma4(a, b0, acc0);
      acc1 = wmma4(a, b1, acc1);
    }
  }

  float bias0 = 0.0f, bias1 = 0.0f;
  if (b1p) { bias0  = bias_scale * b1p[n0 + lp];      bias1  = bias_scale * b1p[n0 + 16 + lp]; }
  if (b2p) { bias0 += b2p[n0 + lp];                   bias1 += b2p[n0 + 16 + lp]; }

  float* crow = C + (size_t)(m0 + half * 8) * ldc + n0 + lp;
  #pragma unroll
  for (int r = 0; r < 8; ++r) {
    float v0 = acc0[r] + bias0;
    float v1 = acc1[r] + bias1;
    if (RELU) { v0 = fmaxf(v0, 0.0f); v1 = fmaxf(v1, 0.0f); }
    crow[0]  = v0;
    crow[16] = v1;
    crow += ldc;
  }
}

// ---------------------------------------------------------------------------
// S[b,i,f] = sum_j ReLU(Af[b,j,f] + Bf[b,i,f])   (FH = 256, N = 128)
// One block per (b,i) row; thread f streams the 128 j-rows (L2-resident).
// ---------------------------------------------------------------------------
__global__ void __launch_bounds__(256)
pair_relu_sum(const float* __restrict__ Af, const float* __restrict__ Bf,
              float* __restrict__ S)
{
  const int ri = blockIdx.x;              // b*128 + i
  const int b  = ri >> 7;
  const int f  = threadIdx.x;             // 0..255
  const float bm = Bf[(size_t)ri * 256 + f];
  const float* ap = Af + (size_t)b * 128 * 256 + f;
  float acc = 0.0f;
  #pragma unroll 4
  for (int j = 0; j < 128; ++j)
    acc += fmaxf(ap[(size_t)j * 256] + bm, 0.0f);
  S[(size_t)ri * 256 + f] = acc;
}

// ---------------------------------------------------------------------------
// Sequential LSTM scan over N=128 steps, B=16, H=128 (4H = 512 gate units).
// Single persistent workgroup; h/c/gates LDS-resident (48 KB). Gx holds the
// precomputed input-path gates: input_g @ wih^T + bih + bhh, laid out (B,N,4H).
// ---------------------------------------------------------------------------
__global__ void __launch_bounds__(512)
lstm_scan(const float* __restrict__ Gx, const float* __restrict__ whh,
          const float* __restrict__ h0, const float* __restrict__ c0,
          float* __restrict__ new_hidden)
{
  __shared__ float h_s[16][128];
  __shared__ float c_s[16][128];
  __shared__ float g_s[16][512];
  const int tid = threadIdx.x;
  for (int idx = tid; idx < 2048; idx += 512) {
    h_s[idx >> 7][idx & 127] = h0[idx];
    c_s[idx >> 7][idx & 127] = c0[idx];
  }
  __syncthreads();
  for (int t = 0; t < 128; ++t) {
    const int u = tid;                    // gate unit 0..511
    float acc[16];
    #pragma unroll
    for (int b = 0; b < 16; ++b)
      acc[b] = Gx[((size_t)(b * 128 + t)) * 512 + u];
    for (int k = 0; k < 128; ++k) {
      const float w = whh[(size_t)u * 128 + k];
      #pragma unroll
      for (int b = 0; b < 16; ++b)
        acc[b] = fmaf(h_s[b][k], w, acc[b]);
    }
    #pragma unroll
    for (int b = 0; b < 16; ++b) g_s[b][u] = acc[b];
    __syncthreads();
    for (int idx = tid; idx < 2048; idx += 512) {
      const int b = idx >> 7, n = idx & 127;
      const float ig = g_s[b][n],       fg = g_s[b][128 + n];
      const float gg = g_s[b][256 + n], og = g_s[b][384 + n];
      const float c = sigmoidf_(fg) * c_s[b][n] + sigmoidf_(ig) * tanhf(gg);
      const float h = sigmoidf_(og) * tanhf(c);
      c_s[b][n] = c; h_s[b][n] = h;
      new_hidden[(size_t)(b * 128 + t) * 128 + n] = h;
    }
    __syncthreads();
  }
}

// ---------------------------------------------------------------------------
extern "C" void kernel_launch(void* const* d_in, const int* in_sizes, int n_in,
                              void* d_out, int out_size, void* d_ws, size_t ws_size,
                              hipStream_t stream)
{
  (void)in_sizes; (void)n_in; (void)out_size; (void)ws_size;
  const float* x      = (const float*)d_in[0];
  const float* hidden = (const float*)d_in[1];
  const float* h0     = (const float*)d_in[2];
  const float* c0     = (const float*)d_in[3];
  const float* f_w1   = (const float*)d_in[4];
  const float* f_b1   = (const float*)d_in[5];
  const float* f_w2   = (const float*)d_in[6];
  const float* f_b2   = (const float*)d_in[7];
  const float* g_w1   = (const float*)d_in[8];
  const float* g_b1   = (const float*)d_in[9];
  const float* g_w2   = (const float*)d_in[10];
  const float* g_b2   = (const float*)d_in[11];
  const float* wih    = (const float*)d_in[12];
  const float* whh    = (const float*)d_in[13];
  const float* bih    = (const float*)d_in[14];
  const float* bhh    = (const float*)d_in[15];
  const float* o_w1   = (const float*)d_in[16];
  const float* o_b1   = (const float*)d_in[17];
  const float* o_w2   = (const float*)d_in[18];
  const float* o_b2   = (const float*)d_in[19];
  float* out = (float*)d_out;

  const int BN = 16 * 128;                    // 2048 row dimension everywhere
  float* R0 = (float*)d_ws;                   // 2048*256 f32
  float* R1 = R0 + (size_t)BN * 256;          // 2048*256
  float* R2 = R1 + (size_t)BN * 256;          // 2048*256
  float* R3 = R2 + (size_t)BN * 256;          // 2048*128 used (new_hidden)

  auto blocks = [](int M, int N) { return ((M >> 4) * (N >> 5) + 7) / 8; };

  // 1) Af = hidden @ f_w1[0:128,:]                         -> R0 (2048x256)
  gemm_wmma_f32<false,false,false><<<blocks(BN,256), 256, 0, stream>>>(
      hidden,128, f_w1,256,128, nullptr,0, nullptr,0,0,
      nullptr,nullptr,1.0f, R0,256, BN,256);
  // 2) Bf = hidden @ f_w1[128:256,:] + f_b1                -> R1 (2048x256)
  gemm_wmma_f32<false,false,false><<<blocks(BN,256), 256, 0, stream>>>(
      hidden,128, f_w1 + 128*256,256,128, nullptr,0, nullptr,0,0,
      f_b1,nullptr,1.0f, R1,256, BN,256);
  // 3) S[b,i] = sum_j ReLU(Af[b,j] + Bf[b,i])              -> R2 (2048x256)
  pair_relu_sum<<<BN, 256, 0, stream>>>(R0, R1, R2);
  // 4) sum_messages = S @ f_w2 + 128*f_b2                  -> R0 (2048x128)
  gemm_wmma_f32<false,false,false><<<blocks(BN,128), 256, 0, stream>>>(
      R2,256, f_w2,128,256, nullptr,0, nullptr,0,0,
      f_b2,nullptr,128.0f, R0,128, BN,128);
  // 5) G1 = ReLU(x @ g_w1[0:128] + sm @ g_w1[128:256] + g_b1) -> R1 (2048x256)
  gemm_wmma_f32<false,true,true><<<blocks(BN,256), 256, 0, stream>>>(
      x,128, g_w1,256,128, R0,128, g_w1 + 128*256,256,128,
      g_b1,nullptr,1.0f, R1,256, BN,256);
  // 6) input_g = G1 @ g_w2 + g_b2                          -> R2 (2048x128)
  gemm_wmma_f32<false,false,false><<<blocks(BN,128), 256, 0, stream>>>(
      R1,256, g_w2,128,256, nullptr,0, nullptr,0,0,
      g_b2,nullptr,1.0f, R2,128, BN,128);
  // 7) Gx = input_g @ wih^T + bih + bhh                    -> R0..R1 (2048x512)
  gemm_wmma_f32<true,false,false><<<blocks(BN,512), 256, 0, stream>>>(
      R2,128, wih,128,128, nullptr,0, nullptr,0,0,
      bih,bhh,1.0f, R0,512, BN,512);
  // 8) LSTM recurrence over N=128 steps                    -> R3 (2048x128)
  lstm_scan<<<1, 512, 0, stream>>>(R0, whh, h0, c0, R3);
  // 9) OH1 = ReLU(new_hidden @ o_w1 + o_b1)                -> R0 (2048x256)
  gemm_wmma_f32<false,false,true><<<blocks(BN,256), 256, 0, stream>>>(
      R3,128, o_w1,256,128, nullptr,0, nullptr,0,0,
      o_b1,nullptr,1.0f, R0,256, BN,256);
  // 10) out = OH1 @ o_w2 + o_b2                            -> d_out (2048x32)
  gemm_wmma_f32<false,false,false><<<blocks(BN,32), 256, 0, stream>>>(
      R0,256, o_w2,32,256, nullptr,0, nullptr,0,0,
      o_b2,nullptr,1.0f, out,32, BN,32);
}